// SpGraphAttentionLayer_21045339750531
// MI455X (gfx1250) — compile-verified
//
#include <hip/hip_runtime.h>
#include <cstdint>
#include <cstddef>

#define N_NODES 50000
#define N_EDGES 800000
#define F 256

typedef float v2f __attribute__((ext_vector_type(2)));
typedef float v8f __attribute__((ext_vector_type(8)));

// ---------------------------------------------------------------------------
// Kernel 1: support = x @ W  (fp32 WMMA, V_WMMA_F32_16X16X4_F32)
// Block: 256 threads = 8 waves; block tile 64(M) x 32(N); K staged in 32-chunks
// through LDS. Each wave computes one 16x16 fp32 tile (8-VGPR accumulator).
// ---------------------------------------------------------------------------
#define BM 64
#define BN 32
#define KB 32

__global__ __launch_bounds__(256) void gemm_xw_wmma(const float* __restrict__ x,
                                                    const float* __restrict__ W,
                                                    float* __restrict__ support) {
    __shared__ float As[BM][KB + 4];   // +4 pad: keeps rows 16B-aligned, avoids bank conflicts
    __shared__ float Bs[KB][BN + 4];

    const int tid   = threadIdx.x;
    const int wid   = tid >> 5;
    const int lane  = tid & 31;
    const int half  = lane >> 4;   // 0: lanes 0-15 (K=0,1), 1: lanes 16-31 (K=2,3)
    const int l16   = lane & 15;
    const int waveM = wid & 3;     // 4 wave-rows
    const int waveN = wid >> 2;    // 2 wave-cols
    const int blockM = blockIdx.x * BM;
    const int blockN = blockIdx.y * BN;

    v8f acc = {};

    for (int k0 = 0; k0 < F; k0 += KB) {
        // ---- stage A tile: 64x32 floats = 512 float4, 2 per thread ----
        #pragma unroll
        for (int i = 0; i < 2; ++i) {
            int linear = tid + i * 256;          // 0..511
            int r  = linear >> 3;                // row 0..63
            int c4 = (linear & 7) << 2;          // col 0,4,..,28
            int gr = blockM + r;
            float4 v = make_float4(0.f, 0.f, 0.f, 0.f);
            if (gr < N_NODES)
                v = *reinterpret_cast<const float4*>(&x[(size_t)gr * F + k0 + c4]);
            *reinterpret_cast<float4*>(&As[r][c4]) = v;
        }
        // ---- stage B tile: 32x32 floats = 256 float4, 1 per thread ----
        {
            int r  = tid >> 3;
            int c4 = (tid & 7) << 2;
            float4 v = *reinterpret_cast<const float4*>(&W[(size_t)(k0 + r) * F + blockN + c4]);
            *reinterpret_cast<float4*>(&Bs[r][c4]) = v;
        }
        __syncthreads();

        const int rA = waveM * 16 + l16;
        const int cB = waveN * 16 + l16;
        #pragma unroll
        for (int kk = 0; kk < KB; kk += 4) {
            // A 16x4 fragment: lane<16 -> K=kk+0,kk+1 ; lane>=16 -> K=kk+2,kk+3
            v2f a, b;
            a.x = As[rA][kk + 2 * half + 0];
            a.y = As[rA][kk + 2 * half + 1];
            // B 4x16 fragment (mirrored K split across lane halves)
            b.x = Bs[kk + 2 * half + 0][cB];
            b.y = Bs[kk + 2 * half + 1][cB];
            acc = __builtin_amdgcn_wmma_f32_16x16x4_f32(
                /*neg_a=*/false, a, /*neg_b=*/false, b,
                /*c_mod=*/(short)0, acc, /*reuse_a=*/false, /*reuse_b=*/false);
        }
        __syncthreads();
    }

    // ---- store 16x16 f32 tile: VGPR v -> M=v (lanes 0-15) / M=v+8 (lanes 16-31)
    const int col = blockN + waveN * 16 + l16;
    #pragma unroll
    for (int v = 0; v < 8; ++v) {
        int row = blockM + waveM * 16 + v + 8 * half;
        if (row < N_NODES)
            support[(size_t)row * F + col] = acc[v];
    }
}

// ---------------------------------------------------------------------------
// Kernel 2: per-node gate.  r = sigmoid(dot(support_row, a2));
// gated_row = r * support_row (in place).  One wave (32 lanes) per node.
// ---------------------------------------------------------------------------
__global__ __launch_bounds__(256) void gate_nodes(float* __restrict__ gated,
                                                  const float* __restrict__ a2,
                                                  float* __restrict__ rD) {
    const int wid  = threadIdx.x >> 5;
    const int lane = threadIdx.x & 31;
    const int node = blockIdx.x * 8 + wid;
    if (node >= N_NODES) return;

    float s[8];
    float dot = 0.f;
    #pragma unroll
    for (int j = 0; j < 8; ++j) {
        int f = lane + 32 * j;                 // coalesced across lanes
        s[j] = gated[(size_t)node * F + f];
        dot += s[j] * a2[f];
    }
    #pragma unroll
    for (int off = 16; off > 0; off >>= 1)
        dot += __shfl_xor(dot, off, 32);

    float r = 1.f / (1.f + expf(-dot));        // sigmoid
    #pragma unroll
    for (int j = 0; j < 8; ++j)
        gated[(size_t)node * F + lane + 32 * j] = r * s[j];
    if (lane == 0) rD[node] = r;
}

// ---------------------------------------------------------------------------
// Kernel 3: edge scatter (segment sums).  One wave per edge.
// out[dst,:] += w * gated[src,:];  sumnorm[dst] += w * rD[src].
// 51.2MB accumulator lives in the 192MB L2 -> hardware fp32 atomics stay on-chip.
// ---------------------------------------------------------------------------
__global__ __launch_bounds__(256) void edge_scatter(const long long* __restrict__ ei,
                                                    const float* __restrict__ ev,
                                                    const float* __restrict__ gated,
                                                    const float* __restrict__ rD,
                                                    float* __restrict__ out,
                                                    float* __restrict__ sumnorm) {
    const int wid  = threadIdx.x >> 5;
    const int lane = threadIdx.x & 31;
    const long long e = (long long)blockIdx.x * 8 + wid;
    if (e >= N_EDGES) return;

    const int dst = (int)ei[e];                // row
    const int src = (int)ei[(long long)N_EDGES + e]; // col
    const float w = ev[e];

    const float* grow = gated + (size_t)src * F;
    float*       orow = out   + (size_t)dst * F;
    #pragma unroll
    for (int j = 0; j < 8; ++j) {
        int f = lane + 32 * j;
        __hip_atomic_fetch_add(&orow[f], w * grow[f],
                               __ATOMIC_RELAXED, __HIP_MEMORY_SCOPE_AGENT);
    }
    if (lane == 0)
        __hip_atomic_fetch_add(&sumnorm[dst], w * rD[src],
                               __ATOMIC_RELAXED, __HIP_MEMORY_SCOPE_AGENT);
}

// ---------------------------------------------------------------------------
// Kernel 4: finalize.  v = out/sumnorm; lD = softplus(dot(v, a1));
// out = elu(max(v,0) + lD*min(v,0)).  One wave per node.
// ---------------------------------------------------------------------------
__global__ __launch_bounds__(256) void finalize_nodes(float* __restrict__ out,
                                                      const float* __restrict__ sumnorm,
                                                      const float* __restrict__ a1) {
    const int wid  = threadIdx.x >> 5;
    const int lane = threadIdx.x & 31;
    const int node = blockIdx.x * 8 + wid;
    if (node >= N_NODES) return;

    const float inv = 1.f / sumnorm[node];
    float v[8];
    float dot = 0.f;
    #pragma unroll
    for (int j = 0; j < 8; ++j) {
        int f = lane + 32 * j;
        v[j] = out[(size_t)node * F + f] * inv;
        dot += v[j] * a1[f];
    }
    #pragma unroll
    for (int off = 16; off > 0; off >>= 1)
        dot += __shfl_xor(dot, off, 32);

    const float lD = (dot > 20.f) ? dot : log1pf(expf(dot));   // softplus
    #pragma unroll
    for (int j = 0; j < 8; ++j) {
        float r = fmaxf(v[j], 0.f) + lD * fminf(v[j], 0.f);
        r = (r > 0.f) ? r : expm1f(r);                          // elu
        out[(size_t)node * F + lane + 32 * j] = r;
    }
}

// ---------------------------------------------------------------------------
extern "C" void kernel_launch(void* const* d_in, const int* in_sizes, int n_in,
                              void* d_out, int out_size, void* d_ws, size_t ws_size,
                              hipStream_t stream) {
    const float*     x  = (const float*)d_in[0];
    const long long* ei = (const long long*)d_in[1];   // int64 edge_index [2,E]
    const float*     ev = (const float*)d_in[2];
    const float*     W  = (const float*)d_in[3];
    const float*     a1 = (const float*)d_in[4];
    const float*     a2 = (const float*)d_in[5];
    float* out = (float*)d_out;

    // workspace layout: gated[N*F] | rD[N] | sumnorm[N]
    float* gated   = (float*)d_ws;
    float* rD      = gated + (size_t)N_NODES * F;
    float* sumnorm = rD + N_NODES;

    hipMemsetAsync(out,     0, (size_t)N_NODES * F * sizeof(float), stream);
    hipMemsetAsync(sumnorm, 0, (size_t)N_NODES * sizeof(float),     stream);

    dim3 g1((N_NODES + BM - 1) / BM, F / BN);
    gemm_xw_wmma<<<g1, 256, 0, stream>>>(x, W, gated);
    gate_nodes<<<(N_NODES + 7) / 8, 256, 0, stream>>>(gated, a2, rD);
    edge_scatter<<<(N_EDGES + 7) / 8, 256, 0, stream>>>(ei, ev, gated, rD, out, sumnorm);
    finalize_nodes<<<(N_NODES + 7) / 8, 256, 0, stream>>>(out, sumnorm, a1);
}